// GIResNetGenerator_51256139710515
// MI455X (gfx1250) — compile-verified
//
#include <hip/hip_runtime.h>

typedef __attribute__((ext_vector_type(2))) float v2f;
typedef __attribute__((ext_vector_type(8))) float v8f;

#define HEADS 8
#define CC    16          // channels per head
#define HH    256
#define WW    256
#define PH    8           // block height  (ph = H/winh)
#define PW    8           // block width
#define P     64          // pixels per block
#define VSTRIDE 68        // padded LDS row stride (floats) to dodge bank conflicts

// One wave32 per (b, head, block i, block j).
__global__ __launch_bounds__(32) void local_mixture_fused(
    const float* __restrict__ logits_read,
    const float* __restrict__ values,
    const float* __restrict__ logits_write,
    float* __restrict__ out)
{
    __shared__ __align__(16) float vbuf[CC * VSTRIDE];   // value tile  [c][p]
    __shared__ __align__(16) float obuf[CC * VSTRIDE];   // output tile [c][p]
    __shared__ __align__(16) float attn[P];
    __shared__ __align__(16) float gate[P];

    const int bx   = blockIdx.x;
    const int j    = bx & 31;
    const int i    = (bx >> 5) & 31;
    const int head = (bx >> 10) & 7;
    const int b    = bx >> 13;
    const int lane = threadIdx.x;

    const size_t img = (size_t)HH * WW;
    const size_t pix_base = (size_t)(i * PH) * WW + (size_t)(j * PW);
    const float* lrp = logits_read  + (size_t)(b * HEADS + head) * img + pix_base;
    const float* lwp = logits_write + (size_t)(b * HEADS + head) * img + pix_base;
    const float* vp  = values + (size_t)((b * HEADS + head) * CC) * img + pix_base;
    float*       op  = out    + (size_t)((b * HEADS + head) * CC) * img + pix_base;

    // ---- softmax over the 64 read-logits (2 per lane: p=lane, p=lane+32) ----
    const int a0 = lane >> 3, col = lane & 7;
    float x0 = lrp[(size_t)a0 * WW + col];
    float x1 = lrp[(size_t)(a0 + 4) * WW + col];
    float m = fmaxf(x0, x1);
    #pragma unroll
    for (int s = 16; s >= 1; s >>= 1) m = fmaxf(m, __shfl_xor(m, s, 32));
    float e0 = __expf(x0 - m), e1 = __expf(x1 - m);
    float ssum = e0 + e1;
    #pragma unroll
    for (int s = 16; s >= 1; s >>= 1) ssum += __shfl_xor(ssum, s, 32);
    const float inv = 1.0f / ssum;
    attn[lane]      = e0 * inv;
    attn[lane + 32] = e1 * inv;

    // ---- sigmoid gates from write-logits ----
    float g0 = lwp[(size_t)a0 * WW + col];
    float g1 = lwp[(size_t)(a0 + 4) * WW + col];
    gate[lane]      = 1.0f / (1.0f + __expf(-g0));
    gate[lane + 32] = 1.0f / (1.0f + __expf(-g1));

    // ---- cooperative load of the 16x64 value tile (float4, coalesced) ----
    #pragma unroll
    for (int k = 0; k < 8; ++k) {
        int f  = k * 32 + lane;          // 0..255 float4 slots
        int c  = f >> 4;
        int a  = (f >> 1) & 7;
        int h4 = f & 1;
        float4 val = *(const float4*)(vp + (size_t)c * img + (size_t)a * WW + h4 * 4);
        *(float4*)&vbuf[c * VSTRIDE + a * 8 + h4 * 4] = val;
    }
    __syncthreads();

    // ---- t[c] = sum_p attn[p] * v[c][p]; lane (c = lane&15, half = lane>>4) ----
    const int c_t  = lane & 15;
    const int half = lane >> 4;
    float acc = 0.0f;
    #pragma unroll
    for (int q = 0; q < 8; ++q) {
        float4 av = *(const float4*)&attn[half * 32 + q * 4];
        float4 vv = *(const float4*)&vbuf[c_t * VSTRIDE + half * 32 + q * 4];
        acc += av.x * vv.x + av.y * vv.y + av.z * vv.z + av.w * vv.w;
    }
    float t = acc + __shfl_xor(acc, 16, 32);   // full 64-term sum in lanes n and n+16

    const bool lo16 = (lane < 16);

    // Prefetch all four gate rows with FULL exec (broadcast reads, no exec masking):
    // lanes 16-31 read the same addresses as lanes 0-15 and are masked by select.
    float gv[4];
    #pragma unroll
    for (int g = 0; g < 4; ++g)
        gv[g] = gate[g * 16 + (lane & 15)];

    // B operand: only K=0 row (lanes 0-15, VGPR0) nonzero -> exact rank-1 product
    v2f Bm;
    Bm.x = lo16 ? t : 0.0f;
    Bm.y = 0.0f;

    // ---- outer product out[p,c] = gate[p] * t[c] via V_WMMA_F32_16X16X4_F32 ----
    const int n_d = lane & 15;
    const int mb  = (lane >> 4) * 8;
    #pragma unroll
    for (int g = 0; g < 4; ++g) {
        v2f Am;
        Am.x = lo16 ? gv[g] : 0.0f;      // A[m][0] = gate[g*16 + m]
        Am.y = 0.0f;
        v8f Cm = {};
        v8f D = __builtin_amdgcn_wmma_f32_16x16x4_f32(
            /*neg_a=*/false, Am, /*neg_b=*/false, Bm,
            /*c_mod=*/(short)0, Cm, /*reuse_a=*/false, /*reuse_b=*/false);
        // D: lane l, VGPR r -> pixel M = r + 8*(l>=16), channel N = l&15
        #pragma unroll
        for (int r = 0; r < 8; ++r)
            obuf[n_d * VSTRIDE + g * 16 + mb + r] = D[r];
    }
    __syncthreads();

    // ---- coalesced float4 stores, mirroring the load pattern ----
    #pragma unroll
    for (int k = 0; k < 8; ++k) {
        int f  = k * 32 + lane;
        int c  = f >> 4;
        int a  = (f >> 1) & 7;
        int h4 = f & 1;
        float4 val = *(const float4*)&obuf[c * VSTRIDE + a * 8 + h4 * 4];
        *(float4*)(op + (size_t)c * img + (size_t)a * WW + h4 * 4) = val;
    }
}

extern "C" void kernel_launch(void* const* d_in, const int* in_sizes, int n_in,
                              void* d_out, int out_size, void* d_ws, size_t ws_size,
                              hipStream_t stream) {
    const float* logits_read  = (const float*)d_in[0];
    const float* values       = (const float*)d_in[1];
    const float* logits_write = (const float*)d_in[2];
    float* out = (float*)d_out;

    const int B = in_sizes[0] / (HEADS * HH * WW);      // = 8
    const int nblocks = B * HEADS * 32 * 32;            // 65536 workgroups
    local_mixture_fused<<<nblocks, 32, 0, stream>>>(logits_read, values, logits_write, out);
}